// GAT_3204045603858
// MI455X (gfx1250) — compile-verified
//
#include <hip/hip_runtime.h>
#include <math.h>

#define HEADS 4
#define NCHID 64
#define NCOUT 64
#define SDIM  32
#define NDIN  128

typedef float v2f __attribute__((ext_vector_type(2)));
typedef float v8f __attribute__((ext_vector_type(8)));

__device__ __forceinline__ float lrelu_(float v)  { return v > 0.f ? v : 0.2f * v; }
__device__ __forceinline__ float relu_(float v)   { return v > 0.f ? v : 0.f; }
__device__ __forceinline__ float sigmoid_(float v){ return 1.f / (1.f + __expf(-v)); }

__device__ __forceinline__ void atomicMaxF(float* addr, float val) {
    unsigned int* ua = reinterpret_cast<unsigned int*>(addr);
    unsigned int cur = __float_as_uint(*addr);
    while (__uint_as_float(cur) < val) {
        unsigned int assumed = cur;
        cur = atomicCAS(ua, assumed, __float_as_uint(val));
        if (cur == assumed) break;
    }
}

__global__ void fill_kernel(float* __restrict__ p, float v, long long n) {
    long long i = (long long)blockIdx.x * blockDim.x + threadIdx.x;
    long long stride = (long long)gridDim.x * blockDim.x;
    for (; i < n; i += stride) p[i] = v;
}

// Out[M,O] = A[M,K] @ W[O,K]^T
// One wave32 per 16(M) x 64(O) strip: 4 independent f32 WMMA accumulators,
// 1 A-fragment load feeds 4 v_wmma_f32_16x16x4_f32 per K-step.
__global__ __launch_bounds__(32)
void gemm_wmma_f32(const float* __restrict__ A, const float* __restrict__ W,
                   float* __restrict__ Out, int M, int K, int O) {
    const int tm   = blockIdx.x * 16;
    const int to   = blockIdx.y * 64;
    const int lane = threadIdx.x & 31;
    const int half = lane >> 4;   // 0: K pair {0,1}, 1: K pair {2,3}
    const int l    = lane & 15;
    int ra = tm + l; if (ra > M - 1) ra = M - 1;          // clamp (no EXEC divergence)
    const float* ar  = A + (size_t)ra * K + 2 * half;
    const float* wr0 = W + (size_t)(to + l) * K + 2 * half;
    const float* wr1 = wr0 + (size_t)16 * K;
    const float* wr2 = wr0 + (size_t)32 * K;
    const float* wr3 = wr0 + (size_t)48 * K;
    v8f acc0 = {}, acc1 = {}, acc2 = {}, acc3 = {};
    for (int k = 0; k < K; k += 4) {
        v2f a;  a.x  = ar[k];  a.y  = ar[k + 1];
        v2f b0; b0.x = wr0[k]; b0.y = wr0[k + 1];
        v2f b1; b1.x = wr1[k]; b1.y = wr1[k + 1];
        v2f b2; b2.x = wr2[k]; b2.y = wr2[k + 1];
        v2f b3; b3.x = wr3[k]; b3.y = wr3[k + 1];
        acc0 = __builtin_amdgcn_wmma_f32_16x16x4_f32(false, a, false, b0, (short)0, acc0, false, false);
        acc1 = __builtin_amdgcn_wmma_f32_16x16x4_f32(false, a, false, b1, (short)0, acc1, false, false);
        acc2 = __builtin_amdgcn_wmma_f32_16x16x4_f32(false, a, false, b2, (short)0, acc2, false, false);
        acc3 = __builtin_amdgcn_wmma_f32_16x16x4_f32(false, a, false, b3, (short)0, acc3, false, false);
    }
    float* orow = Out + (size_t)(tm + half * 8) * O + to + l;
#pragma unroll
    for (int v = 0; v < 8; ++v) {
        int m = tm + half * 8 + v;
        if (m < M) {
            orow[(size_t)v * O]      = acc0[v];
            orow[(size_t)v * O + 16] = acc1[v];
            orow[(size_t)v * O + 32] = acc2[v];
            orow[(size_t)v * O + 48] = acc3[v];
        }
    }
}

// out[n,h] = sum_c h[n, h*C + c] * att[h*C + c]
__global__ void att_score_kernel(const float* __restrict__ h, const float* __restrict__ att,
                                 float* __restrict__ out, int n, int C) {
    int idx = blockIdx.x * blockDim.x + threadIdx.x;
    if (idx >= n * HEADS) return;
    int node = idx >> 2, hh = idx & 3;
    const float* hp = h + (size_t)node * HEADS * C + (size_t)hh * C;
    const float* ap = att + hh * C;
    float s = 0.f;
    for (int c = 0; c < C; ++c) s += hp[c] * ap[c];
    out[idx] = s;
}

// layer-1 softmax max pass over E real edges + n self loops
__global__ void edge_max1_kernel(const int* __restrict__ src, const int* __restrict__ dst,
                                 const float* __restrict__ as_, const float* __restrict__ ad_,
                                 float* __restrict__ mx, int E, int n) {
    long long idx = (long long)blockIdx.x * blockDim.x + threadIdx.x;
    if (idx >= (long long)(E + n) * HEADS) return;
    int e = (int)(idx >> 2), hh = (int)(idx & 3);
    int s = (e < E) ? src[e] : (e - E);
    int d = (e < E) ? dst[e] : (e - E);
    float logit = lrelu_(as_[s * HEADS + hh] + ad_[d * HEADS + hh]);
    atomicMaxF(&mx[d * HEADS + hh], logit);
}

__global__ void edge_exp1_kernel(const int* __restrict__ src, const int* __restrict__ dst,
                                 const float* __restrict__ as_, const float* __restrict__ ad_,
                                 const float* __restrict__ mx, float* __restrict__ ex,
                                 float* __restrict__ sm, int E, int n) {
    long long idx = (long long)blockIdx.x * blockDim.x + threadIdx.x;
    if (idx >= (long long)(E + n) * HEADS) return;
    int e = (int)(idx >> 2), hh = (int)(idx & 3);
    int s = (e < E) ? src[e] : (e - E);
    int d = (e < E) ? dst[e] : (e - E);
    float logit = lrelu_(as_[s * HEADS + hh] + ad_[d * HEADS + hh]);
    float v = __expf(logit - mx[d * HEADS + hh]);
    ex[idx] = v;
    atomicAdd(&sm[d * HEADS + hh], v);
}

// one block (256 threads = H*64 channels) per edge; scatter-add into acc[dst]
__global__ __launch_bounds__(256)
void aggregate_kernel(const int* __restrict__ src, const int* __restrict__ dst,
                      const float* __restrict__ feat, const float* __restrict__ ex,
                      const float* __restrict__ sm, float* __restrict__ acc, int E) {
    int e = blockIdx.x;
    int c = threadIdx.x;               // 0..255
    int hh = c >> 6;
    int s = (e < E) ? src[e] : (e - E);
    int d = (e < E) ? dst[e] : (e - E);
    float coef = ex[(size_t)e * HEADS + hh] / (sm[d * HEADS + hh] + 1e-16f);
    atomicAdd(&acc[(size_t)d * 256 + c], feat[(size_t)s * 256 + c] * coef);
}

// head-mean + bias + relu -> h1 ; block-reduced BN sum/sumsq accumulation
__global__ __launch_bounds__(256)
void h1_bn_stats_kernel(const float* __restrict__ acc, const float* __restrict__ bias1,
                        float* __restrict__ h1, float* __restrict__ stats, int n) {
    __shared__ float s_sum[256], s_sq[256];
    int c   = threadIdx.x & 63;
    int row = threadIdx.x >> 6;   // 0..3
    float lsum = 0.f, lsq = 0.f;
    for (int it = 0; it < 16; ++it) {
        int node = blockIdx.x * 64 + it * 4 + row;
        if (node < n) {
            const float* p = acc + (size_t)node * 256;
            float v = 0.25f * (p[c] + p[64 + c] + p[128 + c] + p[192 + c]) + bias1[c];
            v = relu_(v);
            h1[(size_t)node * 64 + c] = v;
            lsum += v; lsq += v * v;
        }
    }
    s_sum[threadIdx.x] = lsum; s_sq[threadIdx.x] = lsq;
    __syncthreads();
    if (row == 0) {
        float t = s_sum[c] + s_sum[64 + c] + s_sum[128 + c] + s_sum[192 + c];
        float q = s_sq[c]  + s_sq[64 + c]  + s_sq[128 + c]  + s_sq[192 + c];
        atomicAdd(&stats[c], t);
        atomicAdd(&stats[64 + c], q);
    }
}

__global__ void bn_finalize_kernel(float* __restrict__ stats, float inv_n) {
    int c = threadIdx.x;
    if (c < 64) {
        float mu  = stats[c] * inv_n;
        float var = stats[64 + c] * inv_n - mu * mu;
        stats[128 + c] = mu;
        stats[192 + c] = rsqrtf(var + 1e-5f);
    }
}

__global__ void bn_apply_kernel(float* __restrict__ h1, const float* __restrict__ stats, int n) {
    int idx = blockIdx.x * blockDim.x + threadIdx.x;
    if (idx >= n * 64) return;
    int c = idx & 63;
    h1[idx] = (h1[idx] - stats[128 + c]) * stats[192 + c];
}

// per-node: trust = sigmoid(sf.Wt + bt), struct-norm, feature-norm (wave32 per node)
__global__ __launch_bounds__(256)
void node_misc_kernel(const float* __restrict__ sf, const float* __restrict__ x2,
                      const float* __restrict__ Wt, const float* __restrict__ bt,
                      float* __restrict__ trust, float* __restrict__ ns,
                      float* __restrict__ nf, int n) {
    int wid  = (blockIdx.x * blockDim.x + threadIdx.x) >> 5;
    int lane = threadIdx.x & 31;
    if (wid >= n) return;
    float sv = sf[(size_t)wid * SDIM + lane];
    float tdot = sv * Wt[lane];
    float sq = sv * sv;
    float fq = 0.f;
    const float* xr = x2 + (size_t)wid * 256;
#pragma unroll
    for (int t = 0; t < 8; ++t) { float v = xr[lane + 32 * t]; fq += v * v; }
#pragma unroll
    for (int m = 16; m >= 1; m >>= 1) {
        tdot += __shfl_xor(tdot, m, 32);
        sq   += __shfl_xor(sq,   m, 32);
        fq   += __shfl_xor(fq,   m, 32);
    }
    if (lane == 0) {
        trust[wid] = sigmoid_(tdot + bt[0]);
        ns[wid] = fmaxf(sqrtf(sq), 1e-8f);
        nf[wid] = fmaxf(sqrtf(fq), 1e-8f);
    }
}

// wave32 per edge: cosine sims + gate MLP + trust -> per-head logits + atomic max
__global__ __launch_bounds__(256)
void edge_gate_kernel(const int* __restrict__ src, const int* __restrict__ dst,
                      const float* __restrict__ sf, const float* __restrict__ x2,
                      const float* __restrict__ ns, const float* __restrict__ nf,
                      const float* __restrict__ trust,
                      const float* __restrict__ as_, const float* __restrict__ ad_,
                      const float* __restrict__ We1, const float* __restrict__ be1,
                      const float* __restrict__ We2, const float* __restrict__ be2,
                      const float* __restrict__ betas,
                      float* __restrict__ logits, float* __restrict__ mx, int E) {
    int e    = (blockIdx.x * blockDim.x + threadIdx.x) >> 5;
    int lane = threadIdx.x & 31;
    if (e >= E) return;
    int j = src[e], i = dst[e];
    float sd = sf[(size_t)j * SDIM + lane] * sf[(size_t)i * SDIM + lane];
    const float* xj = x2 + (size_t)j * 256;
    const float* xi = x2 + (size_t)i * 256;
    float fd = 0.f;
#pragma unroll
    for (int t = 0; t < 8; ++t) fd += xj[lane + 32 * t] * xi[lane + 32 * t];
#pragma unroll
    for (int m = 16; m >= 1; m >>= 1) {  // butterfly: all lanes get totals
        sd += __shfl_xor(sd, m, 32);
        fd += __shfl_xor(fd, m, 32);
    }
    float ss = sd / (ns[j] * ns[i]);
    float fs = fd / (nf[j] * nf[i]);
    float ag = ss * fs;
    float cf = fabsf(ss - fs);
    float gate = be2[0];
#pragma unroll
    for (int o = 0; o < 8; ++o) {
        float pre = We1[o * 4 + 0] * ss + We1[o * 4 + 1] * fs +
                    We1[o * 4 + 2] * ag + We1[o * 4 + 3] * cf + be1[o];
        gate += We2[o] * relu_(pre);
    }
    float eb = sigmoid_(gate) * relu_(ag);
    float et = sqrtf(fmaxf(trust[j] * trust[i], 0.f));
    float tb = et * relu_(0.5f * (ss + fs));
    float gl = betas[0] * ss + betas[1] * fs + betas[2] * ag + betas[3] * eb + betas[4] * tb;
    if (lane < HEADS) {
        float lg = lrelu_(as_[j * HEADS + lane] + ad_[i * HEADS + lane]) + gl;
        logits[(size_t)e * HEADS + lane] = lg;
        atomicMaxF(&mx[i * HEADS + lane], lg);
    }
}

__global__ void edge_exp2_kernel(const int* __restrict__ dst, const float* __restrict__ mx,
                                 float* __restrict__ lg_ex, float* __restrict__ sm, int E) {
    long long idx = (long long)blockIdx.x * blockDim.x + threadIdx.x;
    if (idx >= (long long)E * HEADS) return;
    int e = (int)(idx >> 2), hh = (int)(idx & 3);
    int i = dst[e];
    float v = __expf(lg_ex[idx] - mx[i * HEADS + hh]);
    lg_ex[idx] = v;
    atomicAdd(&sm[i * HEADS + hh], v);
}

__global__ void finalize2_kernel(const float* __restrict__ acc, const float* __restrict__ b2,
                                 float* __restrict__ out, int n) {
    int idx = blockIdx.x * blockDim.x + threadIdx.x;
    if (idx >= n * NCOUT) return;
    int node = idx >> 6, c = idx & 63;
    const float* p = acc + (size_t)node * 256;
    out[idx] = 0.25f * (p[c] + p[64 + c] + p[128 + c] + p[192 + c]) + b2[c];
}

extern "C" void kernel_launch(void* const* d_in, const int* in_sizes, int n_in,
                              void* d_out, int out_size, void* d_ws, size_t ws_size,
                              hipStream_t stream) {
    const float* x        = (const float*)d_in[0];
    const int*   ei       = (const int*)  d_in[1];
    const float* sf       = (const float*)d_in[2];
    const float* W1       = (const float*)d_in[3];
    const float* att_src1 = (const float*)d_in[4];
    const float* att_dst1 = (const float*)d_in[5];
    const float* bias1    = (const float*)d_in[6];
    const float* W2       = (const float*)d_in[7];
    const float* att_src2 = (const float*)d_in[8];
    const float* att_dst2 = (const float*)d_in[9];
    const float* bias2    = (const float*)d_in[10];
    const float* Wt       = (const float*)d_in[11];
    const float* bt       = (const float*)d_in[12];
    const float* We1      = (const float*)d_in[13];
    const float* be1      = (const float*)d_in[14];
    const float* We2      = (const float*)d_in[15];
    const float* be2      = (const float*)d_in[16];
    const float* betas    = (const float*)d_in[17];

    const int N = in_sizes[0] / NDIN;   // 20000
    const int E = in_sizes[1] / 2;      // 320000
    const int* src  = ei;
    const int* dstp = ei + E;

    // workspace layout (floats), 64-element aligned, with reuse
    float* ws = (float*)d_ws;
    size_t off = 0;
    auto nxt = [&](size_t cnt) { float* p = ws + off; off += (cnt + 63) & ~(size_t)63; return p; };
    float* f_h   = nxt((size_t)N * 256);        // h (layer1), reused as x2 (layer2)
    float* f_acc = nxt((size_t)N * 256);        // out1 accumulator, reused for out2
    float* f_h1  = nxt((size_t)N * 64);
    float* f_as  = nxt((size_t)N * HEADS);
    float* f_ad  = nxt((size_t)N * HEADS);
    float* f_mx  = nxt((size_t)N * HEADS);
    float* f_sm  = nxt((size_t)N * HEADS);
    float* f_ex  = nxt((size_t)(E + N) * HEADS); // exp buf (l1), logits/exp (l2)
    float* f_tr  = nxt((size_t)N);
    float* f_ns  = nxt((size_t)N);
    float* f_nf  = nxt((size_t)N);
    float* f_st  = nxt(256);                     // BN: sum | sumsq | mu | invstd

    const int TB = 256;
    auto blocks = [&](long long n) { return (int)((n + TB - 1) / TB); };

    // ---- layer 1 ----
    fill_kernel<<<1024, TB, 0, stream>>>(f_acc, 0.f, (long long)N * 256);
    fill_kernel<<<blocks((long long)N * HEADS), TB, 0, stream>>>(f_mx, -INFINITY, (long long)N * HEADS);
    fill_kernel<<<blocks((long long)N * HEADS), TB, 0, stream>>>(f_sm, 0.f, (long long)N * HEADS);
    fill_kernel<<<1, TB, 0, stream>>>(f_st, 0.f, 256);

    dim3 g1((N + 15) / 16, (HEADS * NCHID) / 64);
    gemm_wmma_f32<<<g1, 32, 0, stream>>>(x, W1, f_h, N, NDIN, HEADS * NCHID);

    att_score_kernel<<<blocks((long long)N * HEADS), TB, 0, stream>>>(f_h, att_src1, f_as, N, NCHID);
    att_score_kernel<<<blocks((long long)N * HEADS), TB, 0, stream>>>(f_h, att_dst1, f_ad, N, NCHID);

    long long tot1 = (long long)(E + N) * HEADS;
    edge_max1_kernel<<<blocks(tot1), TB, 0, stream>>>(src, dstp, f_as, f_ad, f_mx, E, N);
    edge_exp1_kernel<<<blocks(tot1), TB, 0, stream>>>(src, dstp, f_as, f_ad, f_mx, f_ex, f_sm, E, N);
    aggregate_kernel<<<E + N, 256, 0, stream>>>(src, dstp, f_h, f_ex, f_sm, f_acc, E);

    h1_bn_stats_kernel<<<(N + 63) / 64, 256, 0, stream>>>(f_acc, bias1, f_h1, f_st, N);
    bn_finalize_kernel<<<1, 64, 0, stream>>>(f_st, 1.f / (float)N);
    bn_apply_kernel<<<blocks((long long)N * 64), TB, 0, stream>>>(f_h1, f_st, N);

    // ---- layer 2 ----
    fill_kernel<<<1024, TB, 0, stream>>>(f_acc, 0.f, (long long)N * 256);
    fill_kernel<<<blocks((long long)N * HEADS), TB, 0, stream>>>(f_mx, -INFINITY, (long long)N * HEADS);
    fill_kernel<<<blocks((long long)N * HEADS), TB, 0, stream>>>(f_sm, 0.f, (long long)N * HEADS);

    dim3 g2((N + 15) / 16, (HEADS * NCOUT) / 64);
    gemm_wmma_f32<<<g2, 32, 0, stream>>>(f_h1, W2, f_h, N, NCHID, HEADS * NCOUT);

    att_score_kernel<<<blocks((long long)N * HEADS), TB, 0, stream>>>(f_h, att_src2, f_as, N, NCOUT);
    att_score_kernel<<<blocks((long long)N * HEADS), TB, 0, stream>>>(f_h, att_dst2, f_ad, N, NCOUT);

    node_misc_kernel<<<blocks((long long)N * 32), 256, 0, stream>>>(sf, f_h, Wt, bt, f_tr, f_ns, f_nf, N);
    edge_gate_kernel<<<blocks((long long)E * 32), 256, 0, stream>>>(
        src, dstp, sf, f_h, f_ns, f_nf, f_tr, f_as, f_ad,
        We1, be1, We2, be2, betas, f_ex, f_mx, E);
    edge_exp2_kernel<<<blocks((long long)E * HEADS), TB, 0, stream>>>(dstp, f_mx, f_ex, f_sm, E);
    aggregate_kernel<<<E, 256, 0, stream>>>(src, dstp, f_h, f_ex, f_sm, f_acc, E);
    finalize2_kernel<<<blocks((long long)N * NCOUT), TB, 0, stream>>>(f_acc, bias2, (float*)d_out, N);
}